// LightGCN_16320875725266
// MI455X (gfx1250) — compile-verified
//
#include <hip/hip_runtime.h>

// LightGCN 3-layer propagation for MI455X (gfx1250, wave32).
//  d_out = acc [N_NODES,128] f32 ; d_ws = cur|next ping-pong (2 x 51.2MB),
//  entire working set (153.6MB) resident in the 192MB L2.
//  Scatter layer: 1 wave / edge, float4 gather + 4x global_atomic_add_f32.
//  Accumulate:    exact elementwise add on the WMMA pipe via selector matrices,
//                 fused with zeroing of the dead ping-pong buffer (saves a
//                 full 51MB write pass per layer).

typedef float v8f __attribute__((ext_vector_type(8)));
typedef float v2f __attribute__((ext_vector_type(2)));

#define LATD 128

// ---- init: acc = cur = concat(uEmbeds, iEmbeds); nxt = 0 --------------------
__global__ void lgcn_init_concat(const float4* __restrict__ u,
                                 const float4* __restrict__ it,
                                 float4* __restrict__ acc,
                                 float4* __restrict__ cur,
                                 float4* __restrict__ nxt,
                                 long n_u4, long n_tot4) {
    long t = (long)blockIdx.x * blockDim.x + threadIdx.x;
    if (t >= n_tot4) return;
    float4 v = (t < n_u4) ? u[t] : it[t - n_u4];
    acc[t] = v;
    cur[t] = v;
    nxt[t] = make_float4(0.f, 0.f, 0.f, 0.f);
}

// ---- SpMM scatter: one wave32 per edge --------------------------------------
// lane l handles dims [4l, 4l+4): float4 gather (coalesced 512B/wave), scale,
// 4 hardware f32 atomics into the destination row (stays hot in L2).
__global__ void lgcn_spmm_scatter(const int*   __restrict__ esrc,
                                  const int*   __restrict__ edst,
                                  const float* __restrict__ eval,
                                  const float* __restrict__ cur,
                                  float*       __restrict__ nxt,
                                  int n_edges) {
    int wid  = blockIdx.x * (blockDim.x >> 5) + (threadIdx.x >> 5);
    if (wid >= n_edges) return;                 // wave-uniform
    int lane = threadIdx.x & 31;
    int e = __builtin_amdgcn_readfirstlane(wid);  // SGPR -> s_load of edge data

    int   s = esrc[e];
    int   d = edst[e];
    float w = eval[e];

    const float4* row = reinterpret_cast<const float4*>(cur + (long)s * LATD);
    float4 m = row[lane];

    float* out = nxt + (long)d * LATD + lane * 4;
    unsafeAtomicAdd(out + 0, m.x * w);
    unsafeAtomicAdd(out + 1, m.y * w);
    unsafeAtomicAdd(out + 2, m.z * w);
    unsafeAtomicAdd(out + 3, m.w * w);
}

// ---- acc += next (exact, WMMA pipe) + zero the dead buffer ------------------
// One wave per 16x16 f32 tile. Four chained V_WMMA_F32_16X16X4_F32:
//   A_b[m][k] = (m == 4b+k) ? 1 : 0   (computed, no loads)
//   B_b       = next[r0+4b : r0+4b+4, c0:c0+16]
//   C chain   = acc tile  ->  D = next_tile + acc_tile   (bit-exact in f32)
// Layouts per CDNA5 ISA 7.12.2:
//   A (16x4 f32, 2 VGPR): A[v][lane] = A[m = lane%16][k = v + 2*(lane/16)]
//   B (4x16 f32, 2 VGPR): B[v][lane] = B[k = v + 2*(lane/16)][n = lane%16]
//   C/D (16x16, 8 VGPR): C[v][lane] = T[m = v + 8*(lane/16)][n = lane%16]
// `dead` (the old cur buffer) is zeroed at the same tile addresses so the
// next layer's scatter target is ready without a separate zero pass.
__global__ void lgcn_acc_add_wmma(float* __restrict__ acc,
                                  const float* __restrict__ nxt,
                                  float* __restrict__ dead,
                                  int n_waves) {
    int wave = blockIdx.x * (blockDim.x >> 5) + (threadIdx.x >> 5);
    if (wave >= n_waves) return;                // wave-uniform, EXEC stays all-1s
    int lane   = threadIdx.x & 31;
    int lhalf  = lane >> 4;                     // 0 / 1
    int lm     = lane & 15;

    const int tiles_per_row = LATD / 16;        // 8
    int tr = wave / tiles_per_row;
    int tc = wave % tiles_per_row;
    long r0 = (long)tr * 16;
    int  c0 = tc * 16;

    v8f c;
#pragma unroll
    for (int v = 0; v < 8; ++v)
        c[v] = acc[(r0 + v + 8 * lhalf) * LATD + c0 + lm];

#pragma unroll
    for (int b = 0; b < 4; ++b) {
        v2f a, bb;
#pragma unroll
        for (int v = 0; v < 2; ++v) {
            int k = v + 2 * lhalf;                           // 0..3
            a[v]  = (lm == (4 * b + k)) ? 1.0f : 0.0f;        // selector
            bb[v] = nxt[(r0 + 4 * b + k) * LATD + c0 + lm];   // coalesced row segs
        }
        c = __builtin_amdgcn_wmma_f32_16x16x4_f32(
                /*neg_a=*/false, a, /*neg_b=*/false, bb,
                /*c_mod=*/(short)0, c, /*reuse_a=*/false, /*reuse_b=*/false);
    }

#pragma unroll
    for (int v = 0; v < 8; ++v) {
        long idx = (r0 + v + 8 * lhalf) * LATD + c0 + lm;
        acc[idx]  = c[v];
        dead[idx] = 0.0f;    // prepare old-cur as next layer's scatter target
    }
}

extern "C" void kernel_launch(void* const* d_in, const int* in_sizes, int n_in,
                              void* d_out, int out_size, void* d_ws, size_t ws_size,
                              hipStream_t stream) {
    const float* uEmb = (const float*)d_in[0];
    const float* iEmb = (const float*)d_in[1];
    const int*   esrc = (const int*)d_in[2];
    const int*   edst = (const int*)d_in[3];
    const float* eval = (const float*)d_in[4];

    const long nU = (long)in_sizes[0] / LATD;
    const long nI = (long)in_sizes[1] / LATD;
    const long N  = nU + nI;
    const int  E  = in_sizes[2];
    const long elems = N * LATD;
    const long tot4  = elems / 4;

    if (ws_size < (size_t)(2 * elems) * sizeof(float)) return;

    float* acc = (float*)d_out;
    float* cur = (float*)d_ws;
    float* nxt = cur + elems;

    const int TPB = 256;                     // 8 wave32 per block

    lgcn_init_concat<<<(int)((tot4 + TPB - 1) / TPB), TPB, 0, stream>>>(
        (const float4*)uEmb, (const float4*)iEmb,
        (float4*)acc, (float4*)cur, (float4*)nxt,
        (long)in_sizes[0] / 4, tot4);

    const int waves_acc = (int)((N / 16) * (LATD / 16));

    for (int layer = 0; layer < 3; ++layer) {
        lgcn_spmm_scatter<<<(E + 7) / 8, TPB, 0, stream>>>(
            esrc, edst, eval, cur, nxt, E);

        // acc += nxt via WMMA; simultaneously zero `cur` (dead after this)
        lgcn_acc_add_wmma<<<(waves_acc + 7) / 8, TPB, 0, stream>>>(
            acc, nxt, cur, waves_acc);

        float* t = cur; cur = nxt; nxt = t;  // ping-pong (nxt now pre-zeroed)
    }
}